// Model_26792005992701
// MI455X (gfx1250) — compile-verified
//
#include <hip/hip_runtime.h>
#include <hip/hip_bf16.h>

typedef __attribute__((ext_vector_type(2))) float v2f;
typedef __attribute__((ext_vector_type(8))) float v8f;
typedef __attribute__((ext_vector_type(4))) float f4;

#define SEQ_LEN 512
#define PRED_LEN 192
#define ENC_IN 512
#define BATCH 128

// ---------------------------------------------------------------------------
// Phase 1: per (batch, 16-channel tile) wave computes
//   Sy[n]  = sum_l x[b,l,n]
//   Sty[n] = sum_l t_l * x[b,l,n],  t_l = l - 512
// via V_WMMA_F32_16X16X4_F32:  D(16x16) += A(16x4) * B(4x16)
//   A row 0 = ones, row 1 = t values, rows 2..15 = 0
//   B[k, n] = x[b, l0+k, n0+n]
// VGPR layouts (CDNA5 ISA):
//   A: lanes 0-15 hold M=lane, K={0,1} in vgpr{0,1}; lanes 16-31: K={2,3}
//   B: vgpr0 = rows K=0 (lanes 0-15) / K=2 (lanes 16-31); vgpr1 = K=1 / K=3
//   D: vgpr v, lanes 0-15 = row M=v  -> acc[0]=Sy row, acc[1]=Sty row
// A is carried as a running value: row 1 advances by exactly +4.0 per K-step
// (exact in f32, |t| <= 512), rows 0 and 2..15 advance by 0. This keeps the
// steady-state loop at 2 NT loads + 2 v_add_f32 + 1 v_wmma.
// ---------------------------------------------------------------------------
__global__ __launch_bounds__(256) void trend_fit_wmma(const float* __restrict__ x,
                                                      float* __restrict__ coef) {
    const int lane  = threadIdx.x & 31;
    const int wave  = threadIdx.x >> 5;
    const int gw    = blockIdx.x * 8 + wave;   // 0 .. 4095
    const int b     = gw >> 5;                 // 0 .. 127
    const int ntile = gw & 31;                 // 0 .. 31
    const int n0    = ntile * 16;

    const int m     = lane & 15;               // A-matrix row / D column
    const int khalf = (lane >> 4) << 1;        // 0 for lanes 0-15, 2 for 16-31

    // B operand source: channel n0+m, rows l0+khalf (vgpr0) and l0+khalf+1 (vgpr1)
    const float* p = x + (size_t)b * SEQ_LEN * ENC_IN + (size_t)khalf * ENC_IN + n0 + m;

    // Running A operand (exact integer arithmetic in f32)
    const float t0   = (float)khalf - 512.0f;          // t at l0=0 for this lane half
    v2f a;
    a.x = (m == 0) ? 1.0f : ((m == 1) ? t0        : 0.0f);
    a.y = (m == 0) ? 1.0f : ((m == 1) ? t0 + 1.0f : 0.0f);
    const float astep = (m == 1) ? 4.0f : 0.0f;        // per-K-step advance

    v8f acc = {};
    for (int l0 = 0; l0 < SEQ_LEN; l0 += 4) {
        v2f bb;
        bb.x = __builtin_nontemporal_load(&p[0]);
        bb.y = __builtin_nontemporal_load(&p[ENC_IN]);
        p += 4 * ENC_IN;

        // (neg_a, A, neg_b, B, c_mod, C, reuse_a, reuse_b)
        acc = __builtin_amdgcn_wmma_f32_16x16x4_f32(false, a, false, bb,
                                                    (short)0, acc, false, false);
        a.x += astep;
        a.y += astep;
    }

    if (lane < 16) {
        const float Sy  = acc[0];   // D[0, lane]
        const float Sty = acc[1];   // D[1, lane]
        const float S0 = 512.0f;
        const float S1 = (float)(-131328.0);          // sum t
        const float S2 = (float)(44870400.0);         // sum t^2
        const float inv_det = (float)(1.0 / 5726601216.0);
        const float intercept = (S2 * Sy - S1 * Sty) * inv_det;
        const float slope     = (S0 * Sty - S1 * Sy) * inv_det;
        const int idx = b * ENC_IN + n0 + lane;
        coef[idx]                   = intercept;      // regular store: re-read by phase 2
        coef[BATCH * ENC_IN + idx]  = slope;
    }
}

// ---------------------------------------------------------------------------
// Phase 2: broadcast line over the last PRED_LEN steps (float4 per thread).
// d_out = [dec_out | t_out | s_out], each BATCH*PRED_LEN*ENC_IN floats.
// Output is write-only (151 MB): non-temporal stores keep it out of L2.
// ---------------------------------------------------------------------------
__global__ __launch_bounds__(256) void trend_write(const float* __restrict__ coef,
                                                   float* __restrict__ out) {
    const int NC4 = ENC_IN / 4;                      // 128 float4 per row
    const int NT  = BATCH * PRED_LEN * NC4;          // 3,145,728 threads
    const int idx = blockIdx.x * blockDim.x + threadIdx.x;
    if (idx >= NT) return;

    const int n4 = idx % NC4;
    const int bp = idx / NC4;
    const int p  = bp % PRED_LEN;
    const int b  = bp / PRED_LEN;
    const float tn = (float)(SEQ_LEN - PRED_LEN + p);   // 320 .. 511

    const f4* ic = (const f4*)coef + (size_t)b * NC4 + n4;
    const f4* sl = (const f4*)(coef + BATCH * ENC_IN) + (size_t)b * NC4 + n4;
    const f4 i4 = *ic;          // coef table is hot in L2 (512 KB), regular loads
    const f4 s4 = *sl;

    f4 v;
    v.x = fmaf(s4.x, tn, i4.x);
    v.y = fmaf(s4.y, tn, i4.y);
    v.z = fmaf(s4.z, tn, i4.z);
    v.w = fmaf(s4.w, tn, i4.w);

    const size_t OFF4 = (size_t)BATCH * PRED_LEN * NC4;  // region size in float4
    f4* o = (f4*)out;
    f4 z = {0.0f, 0.0f, 0.0f, 0.0f};
    __builtin_nontemporal_store(v, &o[idx]);             // dec_out
    __builtin_nontemporal_store(v, &o[idx + OFF4]);      // t_out
    __builtin_nontemporal_store(z, &o[idx + 2 * OFF4]);  // s_out
}

extern "C" void kernel_launch(void* const* d_in, const int* in_sizes, int n_in,
                              void* d_out, int out_size, void* d_ws, size_t ws_size,
                              hipStream_t stream) {
    const float* x_enc = (const float*)d_in[0];   // (128, 512, 512) f32
    float* coef = (float*)d_ws;                   // 2 * 128 * 512 floats = 512 KB
    float* out  = (float*)d_out;

    // Phase 1: 4096 waves, 8 waves (256 threads) per block -> 512 blocks
    trend_fit_wmma<<<512, 256, 0, stream>>>(x_enc, coef);

    // Phase 2: 3,145,728 threads / 256 -> 12288 blocks
    const int NT = BATCH * PRED_LEN * (ENC_IN / 4);
    trend_write<<<(NT + 255) / 256, 256, 0, stream>>>(coef, out);
}